// CrossSNRLoss_8383776162604
// MI455X (gfx1250) — compile-verified
//
#include <hip/hip_runtime.h>
#include <hip/hip_bf16.h>

// ---- problem constants (from reference) ----
#define BB    8192
#define TT    2048
#define FF    110          // HIGH-LOW
#define LOWB  40
#define NPAD  224          // 220 cols (sin|cos) padded to 14*16
#define NT    14           // N tiles of 16
#define PI_F  3.14159265f
#define FPS_F 30.0f

typedef float v2f __attribute__((ext_vector_type(2)));
typedef float v8f __attribute__((ext_vector_type(8)));

// ---------------------------------------------------------------------------
// Kernel 1: fused table  W[k][n] = hanning[k] * {sin|cos}(2*pi*((LOW+bin)/60)/FPS * k)
// ---------------------------------------------------------------------------
__global__ void build_tables_kernel(const float* __restrict__ hanning,
                                    float* __restrict__ W) {
    int i = blockIdx.x * blockDim.x + threadIdx.x;     // i < TT*NPAD
    int k = i / NPAD;
    int n = i - k * NPAD;
    float v = 0.0f;
    if (n < 2 * FF) {
        int bin = (n < FF) ? n : (n - FF);
        float w   = (2.0f * PI_F) * (((float)(LOWB + bin)) / 60.0f) / FPS_F;
        float ang = w * (float)k;
        float bfun = (n < FF) ? __sinf(ang) : __cosf(ang);
        v = hanning[k] * bfun;
    }
    W[i] = v;
}

// ---------------------------------------------------------------------------
// Kernel 2: G[B][NPAD] = bvp[B][T] @ W[T][NPAD] via V_WMMA_F32_16X16X4_F32.
// One wave per 16x16 C tile. fp32 A layout: lane<16 holds K={0,1}, lane>=16
// holds K={2,3}, lane%16 = M row. fp32 B layout mirrored: lane%16 = N col.
// ---------------------------------------------------------------------------
__global__ void __launch_bounds__(128)
dft_gemm_kernel(const float* __restrict__ A,   // bvp [BB][TT]
                const float* __restrict__ W,   // [TT][NPAD]
                float* __restrict__ G) {       // [BB][NPAD]
    int wave  = (blockIdx.x * blockDim.x + threadIdx.x) >> 5;
    int lane  = threadIdx.x & 31;
    int mTile = wave / NT;
    int nTile = wave - mTile * NT;

    int r = lane & 15;        // M row (A) / N col (B)
    int h = lane >> 4;        // K half-select

    const float* pA = A + (size_t)(mTile * 16 + r) * TT + 2 * h;     // A[m][2h + k0]
    const float* pB = W + (size_t)(2 * h) * NPAD + nTile * 16 + r;   // W[2h + k0][n]

    v8f acc = {};
#pragma unroll 8
    for (int k0 = 0; k0 < TT; k0 += 4) {
        v2f a = *(const v2f*)(pA + k0);                // A[m][k0+2h], A[m][k0+2h+1]
        v2f b;
        b.x = pB[(size_t)k0 * NPAD];                   // W[k0+2h    ][n]
        b.y = pB[(size_t)(k0 + 1) * NPAD];             // W[k0+2h + 1][n]
        acc = __builtin_amdgcn_wmma_f32_16x16x4_f32(
                  /*neg_a=*/false, a, /*neg_b=*/false, b,
                  /*c_mod=*/(short)0, acc,
                  /*reuse_a=*/false, /*reuse_b=*/false);
    }

    // C/D layout: VGPR j holds M = j + 8*h, lane%16 = N
    float* pC = G + (size_t)(mTile * 16 + 8 * h) * NPAD + nTile * 16 + r;
#pragma unroll
    for (int j = 0; j < 8; ++j) pC[(size_t)j * NPAD] = acc[j];
}

// ---------------------------------------------------------------------------
// Kernel 3: one wave per row. Power spectrum, argmax (first-max), row loss.
// ---------------------------------------------------------------------------
__global__ void __launch_bounds__(256)
row_loss_kernel(const float* __restrict__ G,
                const int*   __restrict__ hr,
                float* __restrict__ rowloss,
                float* __restrict__ outIdx) {
    int lane = threadIdx.x & 31;
    int b    = blockIdx.x * (blockDim.x >> 5) + (threadIdx.x >> 5);
    const float* g = G + (size_t)b * NPAD;

    float ca[4];
    float lsum = 0.0f;
    float lmax = -1.0f;           // ca >= 0, so any bin beats this
    int   lidx = 0x7fffffff;
#pragma unroll
    for (int j = 0; j < 4; ++j) {
        int f = lane + 32 * j;
        float v = 0.0f;
        if (f < FF) {
            float s = g[f];
            float c = g[f + FF];
            v = s * s + c * c;
            lsum += v;
            if (v > lmax) { lmax = v; lidx = f; }
        }
        ca[j] = v;
    }
    // wave32 reductions
#pragma unroll
    for (int off = 16; off > 0; off >>= 1) {
        lsum += __shfl_xor(lsum, off, 32);
        float om = __shfl_xor(lmax, off, 32);
        int   oi = __shfl_xor(lidx, off, 32);
        if (om > lmax || (om == lmax && oi < lidx)) { lmax = om; lidx = oi; }
    }
    float inv = 1.0f / lsum;

    // target / mask window (exclusive right, like reference)
    int h = hr[b];
    if (h >= LOWB + FF) h = LOWB + FF - 1;
    if (h <= LOWB)      h = LOWB;
    int t  = h - LOWB;
    int il = t - 3; if (il < 0) il = 0;
    int ir = t + 3; if (ir > FF - 1) ir = FF - 1;

    float lse = 0.0f, lmask = 0.0f, ltgt = 0.0f;
#pragma unroll
    for (int j = 0; j < 4; ++j) {
        int f = lane + 32 * j;
        if (f < FF) {
            float cn = ca[j] * inv;              // ca_norm in [0,1] -> softmax is stable
            lse += expf(cn);
            if (f >= il && f < ir) lmask += cn;
            if (f == t) ltgt += cn;
        }
    }
#pragma unroll
    for (int off = 16; off > 0; off >>= 1) {
        lse   += __shfl_xor(lse,   off, 32);
        lmask += __shfl_xor(lmask, off, 32);
        ltgt  += __shfl_xor(ltgt,  off, 32);
    }

    if (lane == 0) {
        float logp_t = ltgt - logf(lse);         // log_softmax at target
        rowloss[b]   = -logp_t + (1.0f - lmask); // CE contrib + SNR contrib
        outIdx[b]    = (float)(lidx + LOWB);     // whole_max_idx
    }
}

// ---------------------------------------------------------------------------
// Kernel 4: deterministic fixed-order reduction of row losses -> loss scalar
// ---------------------------------------------------------------------------
__global__ void __launch_bounds__(256)
final_reduce_kernel(const float* __restrict__ rowloss, float* __restrict__ out) {
    __shared__ float sm[256];
    float s = 0.0f;
    for (int i = threadIdx.x; i < BB; i += 256) s += rowloss[i];
    sm[threadIdx.x] = s;
    __syncthreads();
    for (int off = 128; off > 0; off >>= 1) {
        if ((int)threadIdx.x < off) sm[threadIdx.x] += sm[threadIdx.x + off];
        __syncthreads();
    }
    if (threadIdx.x == 0) out[0] = sm[0] / (float)BB;
}

// ---------------------------------------------------------------------------
extern "C" void kernel_launch(void* const* d_in, const int* in_sizes, int n_in,
                              void* d_out, int out_size, void* d_ws, size_t ws_size,
                              hipStream_t stream) {
    const float* bvp     = (const float*)d_in[0];   // [BB*TT] f32
    const float* hanning = (const float*)d_in[1];   // [TT]    f32
    const int*   hr      = (const int*)  d_in[2];   // [BB]    i32
    float* out = (float*)d_out;                     // [0]=loss, [1..BB]=idx

    // workspace layout (floats)
    float* W       = (float*)d_ws;                  // TT*NPAD
    float* G       = W + (size_t)TT * NPAD;         // BB*NPAD
    float* rowloss = G + (size_t)BB * NPAD;         // BB

    // 1) fused window * sin/cos basis table
    build_tables_kernel<<<(TT * NPAD) / 256, 256, 0, stream>>>(hanning, W);

    // 2) WMMA GEMM: (BB/16)*(NPAD/16) = 7168 waves, 4 waves/block
    int nwaves = (BB / 16) * NT;
    dft_gemm_kernel<<<(nwaves * 32) / 128, 128, 0, stream>>>(bvp, W, G);

    // 3) per-row spectrum -> argmax + row loss (8 rows per 256-thread block)
    row_loss_kernel<<<BB / 8, 256, 0, stream>>>(G, hr, rowloss, out + 1);

    // 4) deterministic mean
    final_reduce_kernel<<<1, 256, 0, stream>>>(rowloss, out);
}